// MassDamperSpring_87110526697627
// MI455X (gfx1250) — compile-verified
//
#include <hip/hip_runtime.h>

// ---------------------------------------------------------------------------
// MassDamperSpring on MI455X (gfx1250)
//
// Outputs (concatenated in d_out):
//   ydot : 2048 x 256 floats                 (2 MB)   -- tiny
//   J    : 2048 copies of a 256x256 block    (512 MB) -- store-BW bound, ~22us
//
//   Kernel 1: vdot = [ku|kv] @ y via V_WMMA_F32_16X16X4_F32, band-limited to
//             the two tridiagonal stripes (~10 WMMA per 16x16 tile).
//   Kernel 2: build the 256KB Jb block in LDS (fits in 320KB/WGP), then
//             TENSOR_STORE_FROM_LDS (TDM) DMAs it to 8 replicas with NT
//             stores; wave waits on TENSORcnt. Zero HBM read traffic.
// ---------------------------------------------------------------------------

typedef __attribute__((ext_vector_type(2))) float        v2f;
typedef __attribute__((ext_vector_type(8))) float        v8f;
typedef __attribute__((ext_vector_type(4))) unsigned int v4u;
typedef __attribute__((ext_vector_type(8))) int          v8i;
typedef __attribute__((ext_vector_type(4))) int          v4i;

#define HALF   128
#define SIZE   256
#define NROWS  2048              // NCHUNK*NBATCH
#define YDOT_ELEMS (NROWS * SIZE)

// ---------------------------------------------------------------------------
// Kernel 1: ydot.  128 blocks x 256 threads (8 waves).
// Block b owns rows R0 = b*16..+15.  Wave w computes vdot cols N0 = w*16..+15.
// WMMA f32 16x16x4 layouts (ISA 7.12.2):
//   A 16x4 : lane L -> row m = L&15, K = (L>>4)*2 + {0,1}
//   B 4x16 : lane L -> col n = L&15, K = (L>>4)*2 + {0,1}
//   C/D    : lane L -> col n = L&15, rows m = (L>>4)*8 + v, v=0..7
// G = [ku^T ; kv^T] is banded: coefficient of y[:,j] in vdot[:,n] is
//   u-block (j<128):  j==n: -(K[n]+K[n-1])/Ms[n]; j==n+1: K[n]/Ms[n];
//                     j==n-1: K[n-1]/Ms[n]
//   v-block: same with Cs=C*1e-4.   1/Ms = 1e5/M ; Cs/Ms = 10*C/M.
// ---------------------------------------------------------------------------
__global__ void __launch_bounds__(256)
k1_ydot(const float* __restrict__ t, const float* __restrict__ y,
        const float* __restrict__ K, const float* __restrict__ C,
        const float* __restrict__ M, float* __restrict__ out) {
  const int b    = blockIdx.x;          // row tile (0..127)
  const int tid  = threadIdx.x;
  const int wave = tid >> 5;            // col tile (0..7)
  const int lane = tid & 31;
  const int R0   = b * 16;

  // ---- ydot[:, 0:128] = v = y[:, 128:256]  (float4 copy, 16 rows x 32 f4)
  {
    const float4* ys = (const float4*)y;
    float4*       dd = (float4*)out;
    for (int q = tid; q < 512; q += 256) {
      int r  = R0 + (q >> 5);
      int j4 = q & 31;
      dd[r * 64 + j4] = ys[r * 64 + 32 + j4];
    }
  }

  // ---- WMMA tile for vdot
  const int N0    = wave * 16;
  const int n     = N0 + (lane & 15);   // vdot column (0..127)
  const int kboff = (lane >> 4) * 2;
  const int m     = R0 + (lane & 15);   // A row for this lane

  const float invMs = 1e5f / M[n];
  const float csc   = 10.0f / M[n];
  const float kn    = K[n];
  const float knm1  = (n > 0) ? K[n - 1] : 0.0f;
  const float cn    = C[n];
  const float cnm1  = (n > 0) ? C[n - 1] : 0.0f;

  // per-lane banded coefficients (select-driven, no branches in the loop)
  const float dgu  = -(kn + knm1) * invMs;  // j == n
  const float supu = kn * invMs;            // j == n+1
  const float subu = knm1 * invMs;          // j == n-1
  const float dgv  = -(cn + cnm1) * csc;
  const float supv = cn * csc;
  const float subv = cnm1 * csc;

  v8f acc = {};
  if (n == 0) {                              // fold force into accumulator
    const float invMs0 = 1e5f / M[0];
    for (int v = 0; v < 8; ++v) {
      int r = R0 + (lane >> 4) * 8 + v;
      acc[v] = __sinf(t[r]) * invMs0;
    }
  }

  // Band K-range (wave-uniform): j in [N0-1, N0+16], k multiple of 4,
  // clamped to k<=124 so lanes never touch j>=128 (u) / col>=256 (v).
  const int kstart = (N0 > 0) ? ((N0 - 1) & ~3) : 0;
  const int kend   = (N0 + 16 < 124) ? (N0 + 16) : 124;

  // ---- u-block band
  for (int k = kstart; k <= kend; k += 4) {
    const int kb = k + kboff;                // j for this lane (<=127)
    float2 av = *(const float2*)(y + m * SIZE + kb);
    v2f a; a[0] = av.x; a[1] = av.y;
    const int j0 = kb, j1 = kb + 1;
    v2f bm;
    bm[0] = (j0 == n ? dgu : 0.f) + (j0 == n + 1 ? supu : 0.f) + (j0 == n - 1 ? subu : 0.f);
    bm[1] = (j1 == n ? dgu : 0.f) + (j1 == n + 1 ? supu : 0.f) + (j1 == n - 1 ? subu : 0.f);
    acc = __builtin_amdgcn_wmma_f32_16x16x4_f32(
        false, a, false, bm, (short)0, acc, false, false);
  }
  // ---- v-block band (same jj range, columns +128)
  for (int k = kstart; k <= kend; k += 4) {
    const int kb = k + kboff;                // jj for this lane (<=127)
    float2 av = *(const float2*)(y + m * SIZE + HALF + kb);
    v2f a; a[0] = av.x; a[1] = av.y;
    const int j0 = kb, j1 = kb + 1;
    v2f bm;
    bm[0] = (j0 == n ? dgv : 0.f) + (j0 == n + 1 ? supv : 0.f) + (j0 == n - 1 ? subv : 0.f);
    bm[1] = (j1 == n ? dgv : 0.f) + (j1 == n + 1 ? supv : 0.f) + (j1 == n - 1 ? subv : 0.f);
    acc = __builtin_amdgcn_wmma_f32_16x16x4_f32(
        false, a, false, bm, (short)0, acc, false, false);
  }

  const int col = HALF + n;
  for (int v = 0; v < 8; ++v) {
    int r = R0 + (lane >> 4) * 8 + v;
    out[r * SIZE + col] = acc[v];
  }
}

// ---------------------------------------------------------------------------
// Kernel 2: J broadcast. 256 blocks x 256 threads, 256 KB dynamic LDS.
// Build Jb in LDS (zeros + identity + two tridiagonals), then wave 0 issues
// 8 TENSOR_STORE_FROM_LDS ops (one 256x256 f32 tile per replica, TH=NT).
// ---------------------------------------------------------------------------
__global__ void __launch_bounds__(256)
k2_bcast(const float* __restrict__ K, const float* __restrict__ C,
         const float* __restrict__ M, float* __restrict__ out) {
  extern __shared__ float lds[];        // 65536 floats = 256 KB
  const int tid = threadIdx.x;

  // zero-fill via DS B128 stores
  float4  z4 = make_float4(0.f, 0.f, 0.f, 0.f);
  float4* l4 = (float4*)lds;
  for (int i = tid; i < 16384; i += 256) l4[i] = z4;
  __syncthreads();

  if (tid < HALF) {
    const int i = tid;
    const float invM = 1e5f / M[i];     // 1/Ms[i]
    const float csc  = 10.0f / M[i];    // Cs-scale / Ms[i]
    const float ki   = K[i], kim1 = (i > 0) ? K[i - 1] : 0.0f;
    const float ci   = C[i], cim1 = (i > 0) ? C[i - 1] : 0.0f;
    lds[i * SIZE + HALF + i] = 1.0f;                    // top-right identity
    float* row = lds + (HALF + i) * SIZE;
    row[i] = -(ki + kim1) * invM;                       // ku diag
    if (i < HALF - 1) row[i + 1] = ki * invM;           // ku super
    if (i > 0)        row[i - 1] = kim1 * invM;         // ku sub
    row[HALF + i] = -(ci + cim1) * csc;                 // kv diag
    if (i < HALF - 1) row[HALF + i + 1] = ci * csc;     // kv super
    if (i > 0)        row[HALF + i - 1] = cim1 * csc;   // kv sub
  }
  __syncthreads();

  if (tid < 32) {                       // wave 0 only: TDM is EXEC-agnostic
    const unsigned long long jbase =
        (unsigned long long)out + (unsigned long long)YDOT_ELEMS * 4ull;
    for (int rep = 0; rep < 8; ++rep) {
      const unsigned long long ga =
          jbase + ((unsigned long long)(blockIdx.x * 8 + rep)) * (256ull * 256ull * 4ull);
      // D# group 0: count=1 | lds_addr=0 | global_addr | type=2
      v4u g0;
      g0[0] = 1u;
      g0[1] = 0u;
      g0[2] = (unsigned)(ga & 0xFFFFFFFFull);
      g0[3] = (unsigned)((ga >> 32) & 0x01FFFFFFull) | (2u << 30);
      // D# group 1: data_size=4B, tensor 256x256, tile 256x256, stride0=256
      v8i g1;
      g1[0] = 0x00020000;               // wg_mask=0, data_size=2 (4B)
      g1[1] = (int)(256u << 16);        // tensor_dim0 lo16 at [63:48]
      g1[2] = (int)(256u << 16);        // tensor_dim0 hi=0 | tensor_dim1 lo16
      g1[3] = (int)(256u << 16);        // tensor_dim1 hi=0 | tile_dim0=256
      g1[4] = 256;                      // tile_dim1=256, tile_dim2=0 (2D)
      g1[5] = 256;                      // tensor_dim0_stride lo32
      g1[6] = 0;                        // stride0 hi | stride1 lo16 (65536&0xFFFF=0)
      g1[7] = 1;                        // stride1 bits [47:16] = 65536>>16
      // D# group 2/3: higher dims degenerate (dim=1, tiles unused)
      v4i g2; g2[0] = 1; g2[1] = 1; g2[2] = 65536; g2[3] = 0;
      v4i g3; g3[0] = 0; g3[1] = (1 << 16); g3[2] = 0; g3[3] = 0;
      // cpol = 1 -> TH=NT: 512MB written once, never re-read on device;
      // keep it out of the 192MB L2.
#if defined(__clang_major__) && (__clang_major__ >= 23)
      v8i g4 = {};                      // clang-23 therock headers: 6-arg form
      __builtin_amdgcn_tensor_store_from_lds(g0, g1, g2, g3, g4, 1);
#else
      __builtin_amdgcn_tensor_store_from_lds(g0, g1, g2, g3, 1);
#endif
    }
    __builtin_amdgcn_s_wait_tensorcnt(0);
  }
}

// ---------------------------------------------------------------------------
extern "C" void kernel_launch(void* const* d_in, const int* in_sizes, int n_in,
                              void* d_out, int out_size, void* d_ws, size_t ws_size,
                              hipStream_t stream) {
  const float* t = (const float*)d_in[0];   // (32,64)
  const float* y = (const float*)d_in[1];   // (32,64,256)
  const float* K = (const float*)d_in[2];   // (128,)
  const float* C = (const float*)d_in[3];   // (128,)
  const float* M = (const float*)d_in[4];   // (128,)
  float* out = (float*)d_out;               // ydot (524288) ++ J (134217728)

  k1_ydot<<<128, 256, 0, stream>>>(t, y, K, C, M, out);
  k2_bcast<<<256, 256, 256 * 1024, stream>>>(K, C, M, out);
}